// AttConvModule_3367254360200
// MI455X (gfx1250) — compile-verified
//
#include <hip/hip_runtime.h>
#include <math.h>

typedef __attribute__((ext_vector_type(2))) float v2f;
typedef __attribute__((ext_vector_type(8))) float v8f;

#define BB   8
#define CC   512
#define C2C  256
#define DDp  16
#define HH   58
#define WWd  58
#define NP   196      // 4*7*7 pooled positions
#define MP   208      // 196 padded to 13*16
#define KK   256      // channel-pairs dim
#define GSTR 53248    // per-batch padded g/f/h elems (208*256 == 256*208)
#define SSTR 43264    // per-batch S elems (208*208)

#define YP_N   (BB*CC*NP)     // 802816
#define GFH_N  (BB*GSTR)      // 425984 each
#define S_N    (BB*SSTR)      // 346112

// ---------------- zero fill for padded scratch ----------------
__global__ void k_zero(float* __restrict__ p, int n) {
    int i = blockIdx.x * blockDim.x + threadIdx.x;
    if (i < n) p[i] = 0.0f;
}

// ------- fused depthwise conv(1,3,3) + BN + ReLU + maxpool(4,8,8) -------
// one thread per pooled output (b,c,dp,ho,wo); rolling 3-row register buffer
__global__ void k_conv_bn_relu_pool(const float* __restrict__ x,
                                    const float* __restrict__ w1,
                                    const float* __restrict__ b1,
                                    const float* __restrict__ gamma,
                                    const float* __restrict__ beta,
                                    const float* __restrict__ rmean,
                                    const float* __restrict__ rvar,
                                    float* __restrict__ yP) {
    int idx = blockIdx.x * blockDim.x + threadIdx.x;
    if (idx >= BB * CC * 4 * 49) return;
    int wo = idx % 7;  int t = idx / 7;
    int ho = t % 7;    t /= 7;
    int dp = t % 4;    t /= 4;
    int c  = t % CC;   int b = t / CC;

    float W[9];
#pragma unroll
    for (int i = 0; i < 9; ++i) W[i] = w1[c * 9 + i];
    float sc = gamma[c] * rsqrtf(rvar[c] + 1e-5f);
    float sh = (b1[c] - rmean[c]) * sc + beta[c];

    const float* xb = x + (((long)(b * CC + c)) * DDp + dp * 4) * (HH * WWd)
                        + (ho * 8) * WWd + wo * 8;
    float m = 0.0f;   // ReLU floor: pooled max of non-negative values
    for (int dd = 0; dd < 4; ++dd) {
        const float* p = xb + dd * (HH * WWd);
        float r0[10], r1[10], r2[10];
#pragma unroll
        for (int j = 0; j < 10; ++j) { r0[j] = p[j]; r1[j] = p[WWd + j]; }
#pragma unroll
        for (int rr = 0; rr < 8; ++rr) {
#pragma unroll
            for (int j = 0; j < 10; ++j) r2[j] = p[(rr + 2) * WWd + j];
#pragma unroll
            for (int w = 0; w < 8; ++w) {
                float s = r0[w] * W[0] + r0[w+1] * W[1] + r0[w+2] * W[2]
                        + r1[w] * W[3] + r1[w+1] * W[4] + r1[w+2] * W[5]
                        + r2[w] * W[6] + r2[w+1] * W[7] + r2[w+2] * W[8];
                float v = fmaxf(s * sc + sh, 0.0f);
                m = fmaxf(m, v);
            }
#pragma unroll
            for (int j = 0; j < 10; ++j) { r0[j] = r1[j]; r1[j] = r2[j]; }
        }
    }
    yP[idx] = m;
}

// ------- grouped 1x1 convs g/f/h; g laid out as (196x256 row-major) = torch view -------
__global__ void k_ggh(const float* __restrict__ yP,
                      const float* __restrict__ wg,
                      const float* __restrict__ wf,
                      const float* __restrict__ wh,
                      float* __restrict__ gP,   // (B, MP=208, 256) padded, pre-zeroed
                      float* __restrict__ fP,   // (B, 256, MP) padded
                      float* __restrict__ hP) { // (B, 256, MP) padded
    int idx = blockIdx.x * blockDim.x + threadIdx.x;
    if (idx >= BB * C2C * NP) return;
    int n = idx % NP; int t = idx / NP;
    int g = t % C2C;  int b = t / C2C;
    const float* y0 = yP + ((long)(b * CC + 2 * g)) * NP + n;
    float a0 = y0[0], a1 = y0[NP];
    long base = (long)b * GSTR;
    gP[base + g * NP + n]  = a0 * wg[2*g] + a1 * wg[2*g + 1];  // flat c*196+n == gmat view
    fP[base + g * MP + n]  = a0 * wf[2*g] + a1 * wf[2*g + 1];
    hP[base + g * MP + n]  = a0 * wh[2*g] + a1 * wh[2*g + 1];
}

// ------- S = gmat(196x256) @ f(256x196), padded to 208; 1 wave per 16x16 tile -------
__global__ void k_mm1(const float* __restrict__ gP,
                      const float* __restrict__ fP,
                      float* __restrict__ S) {
    int wid  = (blockIdx.x * blockDim.x + threadIdx.x) >> 5;  // 8 waves/block, exact grid
    int lane = threadIdx.x & 31;
    int b  = wid / 169; int t = wid % 169;
    int mt = t / 13;    int nt = t % 13;
    int half = lane >> 4, l16 = lane & 15;

    const float* Arow = gP + (long)b * GSTR + (mt * 16 + l16) * KK;     // A: row l16, k-contig
    const float* Bcol = fP + (long)b * GSTR + nt * 16 + l16;            // B: col l16, row stride MP

    v8f acc = {};
    for (int k0 = 0; k0 < KK; k0 += 4) {
        int ka = k0 + 2 * half;
        v2f a;  a.x = Arow[ka];          a.y = Arow[ka + 1];
        v2f bv; bv.x = Bcol[(long)ka * MP]; bv.y = Bcol[(long)(ka + 1) * MP];
        acc = __builtin_amdgcn_wmma_f32_16x16x4_f32(false, a, false, bv,
                                                    (short)0, acc, false, false);
    }
    float* Sout = S + (long)b * SSTR + (mt * 16 + half * 8) * MP + nt * 16 + l16;
#pragma unroll
    for (int r = 0; r < 8; ++r) Sout[r * MP] = acc[r];
}

// ------- row softmax over 196 valid columns; one wave32 per row -------
__global__ void k_softmax(float* __restrict__ S) {
    int wid  = (blockIdx.x * blockDim.x + threadIdx.x) >> 5;
    int lane = threadIdx.x & 31;
    int b = wid / NP, i = wid % NP;
    float* row = S + (long)b * SSTR + (long)i * MP;

    float v[7];
    float mx = -3.0e38f;
#pragma unroll
    for (int t = 0; t < 7; ++t) {
        int j = lane + 32 * t;
        v[t] = (j < NP) ? row[j] : -3.0e38f;
        mx = fmaxf(mx, v[t]);
    }
#pragma unroll
    for (int o = 16; o > 0; o >>= 1) mx = fmaxf(mx, __shfl_xor(mx, o, 32));
    float sum = 0.0f;
#pragma unroll
    for (int t = 0; t < 7; ++t) {
        int j = lane + 32 * t;
        float e = (j < NP) ? __expf(v[t] - mx) : 0.0f;
        v[t] = e; sum += e;
    }
#pragma unroll
    for (int o = 16; o > 0; o >>= 1) sum += __shfl_xor(sum, o, 32);
    float inv = 1.0f / sum;
#pragma unroll
    for (int t = 0; t < 7; ++t) {
        int j = lane + 32 * t;
        if (j < NP) row[j] = v[t] * inv;
    }
}

// ------- out = h(256x196) @ soft(196x196), K padded to 208; write d_out directly -------
__global__ void k_mm2(const float* __restrict__ hP,
                      const float* __restrict__ S,
                      float* __restrict__ out) {
    int wid  = (blockIdx.x * blockDim.x + threadIdx.x) >> 5;
    int lane = threadIdx.x & 31;
    int b  = wid / 208; int t = wid % 208;
    int mt = t / 13;    int nt = t % 13;
    int half = lane >> 4, l16 = lane & 15;

    const float* Arow = hP + (long)b * GSTR + (mt * 16 + l16) * MP;
    const float* Bcol = S  + (long)b * SSTR + nt * 16 + l16;

    v8f acc = {};
    for (int k0 = 0; k0 < MP; k0 += 4) {
        int ka = k0 + 2 * half;
        v2f a;  a.x = Arow[ka];             a.y = Arow[ka + 1];
        v2f bv; bv.x = Bcol[(long)ka * MP]; bv.y = Bcol[(long)(ka + 1) * MP];
        acc = __builtin_amdgcn_wmma_f32_16x16x4_f32(false, a, false, bv,
                                                    (short)0, acc, false, false);
    }
    int j = nt * 16 + l16;
    if (j < NP) {
        float* o = out + (long)b * (C2C * NP) + (mt * 16 + half * 8) * NP + j;
#pragma unroll
        for (int r = 0; r < 8; ++r) o[r * NP] = acc[r];
    }
}

extern "C" void kernel_launch(void* const* d_in, const int* in_sizes, int n_in,
                              void* d_out, int out_size, void* d_ws, size_t ws_size,
                              hipStream_t stream) {
    const float* x      = (const float*)d_in[0];
    const float* conv1w = (const float*)d_in[1];
    const float* conv1b = (const float*)d_in[2];
    const float* gamma  = (const float*)d_in[3];
    const float* beta   = (const float*)d_in[4];
    const float* rmean  = (const float*)d_in[5];
    const float* rvar   = (const float*)d_in[6];
    const float* wg     = (const float*)d_in[7];
    const float* wf     = (const float*)d_in[8];
    const float* wh     = (const float*)d_in[9];
    float* out = (float*)d_out;

    float* yP = (float*)d_ws;          // 802816
    float* gP = yP + YP_N;             // 425984
    float* fP = gP + GFH_N;            // 425984
    float* hP = fP + GFH_N;            // 425984
    float* S  = hP + GFH_N;            // 346112

    // zero padded g/f/h scratch (contiguous 3*GFH_N floats)
    {
        int n = 3 * GFH_N;
        k_zero<<<(n + 255) / 256, 256, 0, stream>>>(gP, n);
    }
    // stage 1: conv+BN+ReLU+pool  (802816 threads)
    k_conv_bn_relu_pool<<<(BB*CC*4*49 + 255) / 256, 256, 0, stream>>>(
        x, conv1w, conv1b, gamma, beta, rmean, rvar, yP);
    // stage 2: grouped 1x1 convs  (401408 threads)
    k_ggh<<<(BB*C2C*NP + 255) / 256, 256, 0, stream>>>(yP, wg, wf, wh, gP, fP, hP);
    // stage 3a: S = gmat @ f   (8*169 waves, 8 waves/block -> 169 blocks exact)
    k_mm1<<<169, 256, 0, stream>>>(gP, fP, S);
    // stage 3b: softmax rows   (8*196 waves -> 196 blocks exact)
    k_softmax<<<196, 256, 0, stream>>>(S);
    // stage 3c: out = h @ soft (8*208 waves -> 208 blocks exact)
    k_mm2<<<208, 256, 0, stream>>>(hP, S, out);
}